// GCN_44281112821800
// MI455X (gfx1250) — compile-verified
//
#include <hip/hip_runtime.h>
#include <math.h>

#define N_NODES 100000
#define N_EDGES 1600000
#define N_FEATS 64
#define HID     128

typedef float v2f __attribute__((ext_vector_type(2)));
typedef float v8f __attribute__((ext_vector_type(8)));

// ---------------- degree / normalization ----------------
__global__ void k_init_deg(float* __restrict__ deg) {
  int i = blockIdx.x * blockDim.x + threadIdx.x;
  if (i < N_NODES) deg[i] = 1.0f;  // self-loop contributes 1
}

__global__ void k_deg_accum(const int* __restrict__ dst, float* __restrict__ deg) {
  int e = blockIdx.x * blockDim.x + threadIdx.x;
  if (e < N_EDGES) unsafeAtomicAdd(&deg[dst[e]], 1.0f);
}

__global__ void k_dinv(float* __restrict__ deg) {
  int i = blockIdx.x * blockDim.x + threadIdx.x;
  if (i < N_NODES) deg[i] = rsqrtf(deg[i]);  // deg >= 1 always
}

// ---------------- fp32 WMMA GEMM: H[N,Cout] = X[N,Cin] @ W[Cout,Cin]^T ------
// One wave computes one 16x16 tile of H. N_NODES % 16 == 0, Cin % 4 == 0.
// A (16x4 f32): lanes 0-15 hold M=lane, K={0,1}; lanes 16-31 hold M=lane-16, K={2,3}.
// B (4x16 f32): lanes 0-15 hold N=lane, K={0,1}; lanes 16-31 hold N=lane-16, K={2,3}.
// C/D (16x16 f32): VGPR j -> M = j + 8*(lane>=16), N = lane&15.
__global__ void k_gemm_wmma(const float* __restrict__ X, const float* __restrict__ W,
                            float* __restrict__ H, int Cin, int Cout) {
  const int lane = threadIdx.x & 31;
  const int wid  = (blockIdx.x * blockDim.x + threadIdx.x) >> 5;  // wave-uniform
  const int nT    = Cout >> 4;
  const int tiles = (N_NODES / 16) * nT;
  if (wid >= tiles) return;  // whole wave exits together; EXEC stays all-1s for WMMA
  const int mT  = wid / nT;
  const int nTi = wid - mT * nT;
  const int half = lane >> 4;   // selects K pair
  const int l15  = lane & 15;

  const float* Ap = X + (size_t)(mT * 16 + l15) * Cin + half * 2;
  const float* Bp = W + (size_t)(nTi * 16 + l15) * Cin + half * 2;

  v8f acc = {};
  for (int k0 = 0; k0 < Cin; k0 += 4) {
    v2f a = *(const v2f*)(Ap + k0);
    v2f b = *(const v2f*)(Bp + k0);
    acc = __builtin_amdgcn_wmma_f32_16x16x4_f32(false, a, false, b,
                                                (short)0, acc, false, false);
  }

  const int col     = nTi * 16 + l15;
  const int rowBase = mT * 16 + half * 8;
#pragma unroll
  for (int j = 0; j < 8; ++j)
    H[(size_t)(rowBase + j) * Cout + col] = acc[j];
}

// ---------------- zero / scatter / finalize ----------------
__global__ void k_zero4(float4* __restrict__ p, int n4) {
  int t = blockIdx.x * blockDim.x + threadIdx.x;
  if (t < n4) p[t] = make_float4(0.f, 0.f, 0.f, 0.f);
}

// One thread per (edge, 4-feature chunk); consecutive lanes share the edge so
// gathers are coalesced float4s and atomics land on one cacheline cluster.
__global__ void k_agg(const float* __restrict__ H, const int* __restrict__ src,
                      const int* __restrict__ dst, const float* __restrict__ dinv,
                      float* __restrict__ OUT, int Cout, int cshift) {
  const long long total = (long long)N_EDGES << cshift;
  const long long t = (long long)blockIdx.x * blockDim.x + threadIdx.x;
  if (t >= total) return;
  const int e = (int)(t >> cshift);
  const int c = ((int)t & ((1 << cshift) - 1)) << 2;  // feature offset
  const int s = src[e], d = dst[e];
  const float nm = dinv[s] * dinv[d];
  const float4 v = *(const float4*)(H + (size_t)s * Cout + c);
  float* o = OUT + (size_t)d * Cout + c;
  unsafeAtomicAdd(o + 0, v.x * nm);
  unsafeAtomicAdd(o + 1, v.y * nm);
  unsafeAtomicAdd(o + 2, v.z * nm);
  unsafeAtomicAdd(o + 3, v.w * nm);
}

// OUT[i,c] = (relu)(OUT[i,c] + H[i,c]*dinv[i]^2 + b[c])   (self-loop fused here)
__global__ void k_finalize(float* __restrict__ OUT, const float* __restrict__ H,
                           const float* __restrict__ dinv, const float* __restrict__ bias,
                           int fshift, int do_relu) {
  const long long total = (long long)N_NODES << fshift;
  const long long t = (long long)blockIdx.x * blockDim.x + threadIdx.x;
  if (t >= total) return;
  const int node = (int)(t >> fshift);
  const int c    = (int)t & ((1 << fshift) - 1);
  const float di = dinv[node];
  float v = OUT[t] + H[t] * di * di + bias[c];
  if (do_relu) v = fmaxf(v, 0.0f);
  OUT[t] = v;
}

// ---------------- softmax over 64 features, one wave per node ----------------
__global__ void k_softmax64(float* __restrict__ OUT) {
  const int lane = threadIdx.x & 31;
  const int node = (blockIdx.x * blockDim.x + threadIdx.x) >> 5;
  if (node >= N_NODES) return;
  float* row = OUT + (size_t)node * 64;
  float2 v = *(float2*)(row + lane * 2);
  float m = fmaxf(v.x, v.y);
#pragma unroll
  for (int off = 16; off; off >>= 1) m = fmaxf(m, __shfl_xor(m, off, 32));
  const float ex = __expf(v.x - m);
  const float ey = __expf(v.y - m);
  float s = ex + ey;
#pragma unroll
  for (int off = 16; off; off >>= 1) s += __shfl_xor(s, off, 32);
  const float inv = 1.0f / s;
  float2 r; r.x = ex * inv; r.y = ey * inv;
  *(float2*)(row + lane * 2) = r;
}

// ---------------- host orchestration ----------------
static inline void launch_gemm(const float* X, const float* W, float* H,
                               int Cin, int Cout, hipStream_t stream) {
  int tiles  = (N_NODES / 16) * (Cout >> 4);
  int thr    = tiles * 32;
  int blocks = (thr + 255) / 256;
  k_gemm_wmma<<<blocks, 256, 0, stream>>>(X, W, H, Cin, Cout);
}

extern "C" void kernel_launch(void* const* d_in, const int* in_sizes, int n_in,
                              void* d_out, int out_size, void* d_ws, size_t ws_size,
                              hipStream_t stream) {
  const float* x   = (const float*)d_in[0];
  const int*   ei  = (const int*)d_in[1];   // int32 on device (JAX x64 disabled)
  const int*   src = ei;
  const int*   dst = ei + N_EDGES;
  const float* W1  = (const float*)d_in[2];
  const float* b1  = (const float*)d_in[3];
  const float* W2  = (const float*)d_in[4];
  const float* b2  = (const float*)d_in[5];
  const float* W3  = (const float*)d_in[6];
  const float* b3  = (const float*)d_in[7];
  float* out = (float*)d_out;

  // workspace: dinv[N] | bufH[N*128] | bufA[N*128]  (~103 MB)
  float* ws   = (float*)d_ws;
  float* dinv = ws;
  float* bufH = ws + N_NODES;
  float* bufA = bufH + (size_t)N_NODES * HID;

  const int B = 256;

  // degrees -> dinv (in place)
  k_init_deg <<<(N_NODES + B - 1) / B, B, 0, stream>>>(dinv);
  k_deg_accum<<<(N_EDGES + B - 1) / B, B, 0, stream>>>(dst, dinv);
  k_dinv     <<<(N_NODES + B - 1) / B, B, 0, stream>>>(dinv);

  const int n128_4 = N_NODES * HID / 4;          // float4 count for 128-wide buffer
  const int n64_4  = N_NODES * N_FEATS / 4;      // float4 count for 64-wide buffer
  const long long agg128 = (long long)N_EDGES << 5;
  const long long agg64  = (long long)N_EDGES << 4;
  const long long fin128 = (long long)N_NODES << 7;
  const long long fin64  = (long long)N_NODES << 6;

  // ---- layer 1: 64 -> 128, relu ----
  launch_gemm(x, W1, bufH, N_FEATS, HID, stream);
  k_zero4   <<<(n128_4 + B - 1) / B, B, 0, stream>>>((float4*)bufA, n128_4);
  k_agg     <<<(int)((agg128 + B - 1) / B), B, 0, stream>>>(bufH, src, dst, dinv, bufA, HID, 5);
  k_finalize<<<(int)((fin128 + B - 1) / B), B, 0, stream>>>(bufA, bufH, dinv, b1, 7, 1);

  // ---- layer 2: 128 -> 128, relu ----
  launch_gemm(bufA, W2, bufH, HID, HID, stream);
  k_zero4   <<<(n128_4 + B - 1) / B, B, 0, stream>>>((float4*)bufA, n128_4);
  k_agg     <<<(int)((agg128 + B - 1) / B), B, 0, stream>>>(bufH, src, dst, dinv, bufA, HID, 5);
  k_finalize<<<(int)((fin128 + B - 1) / B), B, 0, stream>>>(bufA, bufH, dinv, b2, 7, 1);

  // ---- layer 3: 128 -> 64, no relu; aggregate straight into d_out ----
  launch_gemm(bufA, W3, bufH, HID, N_FEATS, stream);
  k_zero4   <<<(n64_4 + B - 1) / B, B, 0, stream>>>((float4*)out, n64_4);
  k_agg     <<<(int)((agg64 + B - 1) / B), B, 0, stream>>>(bufH, src, dst, dinv, out, N_FEATS, 4);
  k_finalize<<<(int)((fin64 + B - 1) / B), B, 0, stream>>>(out, bufH, dinv, b3, 6, 0);

  // ---- softmax over 64 features ----
  int smThreads = N_NODES * 32;
  k_softmax64<<<(smThreads + B - 1) / B, B, 0, stream>>>(out);
}